// GlobalPointer_36764920053971
// MI455X (gfx1250) — compile-verified
//
#include <hip/hip_runtime.h>

// ---------------------------------------------------------------------------
// GlobalPointer attention-logits for MI455X (gfx1250, wave32, WMMA).
// Stage 1: x = X@W + b, fused RoPE, write Q/K fp32 to workspace.
// Stage 2: logits = rope(Q) . rope(K)^T with pad/tri masking, *1/8.
// Matrix math: V_WMMA_F32_16X16X4_F32 (native fp32 WMMA on CDNA5).
// All LDS fragment reads are contiguous 8B-aligned v2f -> ds_load_*b64,
// no VGPR repacking in the hot loops.
// ---------------------------------------------------------------------------

typedef float v2f __attribute__((ext_vector_type(2)));
typedef float v8f __attribute__((ext_vector_type(8)));

#define NHEADS 12
#define HS     64
#define SEQ    1024
#define DIM    768
#define NCOL   1536   // 2 * NHEADS * HS
#define NBATCH 8

// D = A(16x4) * B(4x16) + C(16x16), fp32.
// (neg_a, A, neg_b, B, c_mod, C, reuse_a, reuse_b)
__device__ __forceinline__ v8f wmma_f32(v2f a, v2f b, v8f c) {
  return __builtin_amdgcn_wmma_f32_16x16x4_f32(
      false, a, false, b, (short)0, c, false, false);
}

// ---------------------------------------------------------------------------
// Kernel 1: QKV projection + bias + RoPE.
// Block = 256 threads (8 waves). Tile: 64 rows (of B*L) x 128 cols (1 head).
// grid.x = 8192/64 = 128 row tiles, grid.y = 12 heads.
// ---------------------------------------------------------------------------
__global__ __launch_bounds__(256)
void qkv_rope_kernel(const float* __restrict__ X,   // [8192][768]
                     const float* __restrict__ W,   // [768][1536]
                     const float* __restrict__ bias,// [1536]
                     float* __restrict__ Qws,       // [B][H][L][64]
                     float* __restrict__ Kws) {     // [B][H][L][64]
  __shared__ float Xs[64][36];   // 64 rows x 32 K (pitch 36: frag reads conflict-free)
  __shared__ float Wt[128][34];  // W tile TRANSPOSED: [n][k], pitch 34 (even -> b64 ok)

  const int rowTile = blockIdx.x;          // 0..127
  const int head    = blockIdx.y;          // 0..11
  const int tid     = threadIdx.x;
  const int lane    = tid & 31;
  const int wave    = tid >> 5;
  const int half    = lane >> 4;           // 0: lanes 0-15, 1: lanes 16-31
  const int lm      = lane & 15;
  const int strip   = wave & 3;            // 16-row strip within 64-row tile
  const int colHalf = wave >> 2;           // 0: q cols, 1: k cols
  const int row0    = rowTile * 64;
  const int col0    = head * 128;

  v8f acc[4];
  #pragma unroll
  for (int ct = 0; ct < 4; ++ct)
    #pragma unroll
    for (int v = 0; v < 8; ++v) acc[ct][v] = 0.0f;

  for (int k0 = 0; k0 < DIM; k0 += 32) {
    // Stage X tile: 64x32 floats, float4-coalesced (2 passes of 256 thr).
    #pragma unroll
    for (int p = 0; p < 2; ++p) {
      const int linear = tid + p * 256;
      const int r = linear >> 3, q4 = linear & 7;
      const float4 v = *(const float4*)(X + (size_t)(row0 + r) * DIM + k0 + q4 * 4);
      *(float4*)&Xs[r][q4 * 4] = v;
    }
    // Stage W tile transposed: coalesced float4 row reads, scalar LDS col writes.
    #pragma unroll
    for (int p = 0; p < 4; ++p) {
      const int linear = tid + p * 256;
      const int kr = linear >> 5, q4 = linear & 31;
      const float4 v = *(const float4*)(W + (size_t)(k0 + kr) * NCOL + col0 + q4 * 4);
      Wt[q4 * 4 + 0][kr] = v.x;
      Wt[q4 * 4 + 1][kr] = v.y;
      Wt[q4 * 4 + 2][kr] = v.z;
      Wt[q4 * 4 + 3][kr] = v.w;
    }
    // Speculative prefetch of next K-chunk into caches (global_prefetch_b8).
    if (k0 + 32 < DIM) {
      __builtin_prefetch(X + (size_t)(row0 + (tid >> 3)) * DIM + k0 + 32, 0, 1);
      __builtin_prefetch(W + (size_t)(k0 + 32 + (tid >> 5)) * NCOL + col0, 0, 1);
    }
    __syncthreads();

    #pragma unroll
    for (int kk = 0; kk < 8; ++kk) {
      const int kb = kk * 4 + 2 * half;  // half-wave picks K pair (0,1)/(2,3)
      const v2f a = *(const v2f*)&Xs[strip * 16 + lm][kb];   // ds_load_b64
      #pragma unroll
      for (int ct = 0; ct < 4; ++ct) {
        const int n = colHalf * 64 + ct * 16 + lm;
        const v2f b = *(const v2f*)&Wt[n][kb];               // ds_load_b64
        acc[ct] = wmma_f32(a, b, acc[ct]);
      }
    }
    __syncthreads();
  }

  // Epilogue: + bias, RoPE (pair partner lives in lane^1), write Q or K.
  float* dst = (colHalf == 0) ? Qws : Kws;
  #pragma unroll
  for (int ct = 0; ct < 4; ++ct) {
    const int d  = ct * 16 + lm;                 // head dim 0..63
    const float bv = bias[col0 + colHalf * 64 + d];
    // inv_freq = 10000^(-2*(d/2)/64) = exp(-(d/2) * 2*ln(10000)/64)
    const float inv = expf(-(float)(d >> 1) * 0.2878231366242558f);
    #pragma unroll
    for (int v = 0; v < 8; ++v) {
      const int row = row0 + strip * 16 + v + 8 * half;  // 0..8191
      const int pos = row & (SEQ - 1);
      const int bi  = row >> 10;
      float s, c;
      sincosf((float)pos * inv, &s, &c);
      const float val     = acc[ct][v] + bv;
      const float partner = __shfl_xor(val, 1, 32);
      const float o = ((d & 1) == 0) ? (val * c - partner * s)
                                     : (val * c + partner * s);
      dst[(((size_t)bi * NHEADS + head) * SEQ + pos) * HS + d] = o;
    }
  }
}

// ---------------------------------------------------------------------------
// Kernel 2: logits[b,h,l,m] = (Q[l,:] . K[m,:]) with masking, /8.
// Block = 256 threads (8 waves). Tile: 128 (l) x 64 (m), K-dim = 64 fully in LDS.
// grid = (16 m-tiles, 8 l-tiles, 96 b*h).
// ---------------------------------------------------------------------------
__global__ __launch_bounds__(256)
void attn_logits_kernel(const float* __restrict__ Qws,
                        const float* __restrict__ Kws,
                        const int*  __restrict__ mask,   // [B][L]
                        float* __restrict__ out) {       // [B*H][L][L]
  __shared__ float Qs[128][68];
  __shared__ float Ks[64][68];

  const int mTile = blockIdx.x;   // 0..15
  const int lTile = blockIdx.y;   // 0..7
  const int bh    = blockIdx.z;   // 0..95
  const int bi    = bh / NHEADS;
  const int tid   = threadIdx.x;
  const int lane  = tid & 31;
  const int wave  = tid >> 5;     // 16-row l-strip
  const int half  = lane >> 4;
  const int lm    = lane & 15;
  const int l0    = lTile * 128;
  const int m0    = mTile * 64;
  const size_t qbase = (size_t)bh * SEQ * HS;

  #pragma unroll
  for (int p = 0; p < 8; ++p) {                 // Q tile 128x64
    const int linear = tid + p * 256;
    const int r = linear >> 4, q4 = linear & 15;
    *(float4*)&Qs[r][q4 * 4] =
        *(const float4*)(Qws + qbase + (size_t)(l0 + r) * HS + q4 * 4);
  }
  #pragma unroll
  for (int p = 0; p < 4; ++p) {                 // K tile 64x64
    const int linear = tid + p * 256;
    const int r = linear >> 4, q4 = linear & 15;
    *(float4*)&Ks[r][q4 * 4] =
        *(const float4*)(Kws + qbase + (size_t)(m0 + r) * HS + q4 * 4);
  }
  __syncthreads();

  v8f acc[4];
  #pragma unroll
  for (int ct = 0; ct < 4; ++ct)
    #pragma unroll
    for (int v = 0; v < 8; ++v) acc[ct][v] = 0.0f;

  #pragma unroll
  for (int ks = 0; ks < 16; ++ks) {
    const int kb = ks * 4 + 2 * half;
    const v2f a = *(const v2f*)&Qs[wave * 16 + lm][kb];      // ds_load_b64
    #pragma unroll
    for (int ct = 0; ct < 4; ++ct) {
      // B[k][n] = K[n][k]: row-major K tile already has k contiguous.
      const v2f b = *(const v2f*)&Ks[ct * 16 + lm][kb];      // ds_load_b64
      acc[ct] = wmma_f32(a, b, acc[ct]);
    }
  }

  // Epilogue: pad mask -> -1e12 ; (l > m) -> subtract 1e12 ; scale 1/8.
  // 403MB output > 192MB L2: non-temporal stores keep Q/K resident in L2.
  #pragma unroll
  for (int ct = 0; ct < 4; ++ct) {
    const int m   = m0 + ct * 16 + lm;
    const int mOk = mask[bi * SEQ + m];
    #pragma unroll
    for (int v = 0; v < 8; ++v) {
      const int l   = l0 + wave * 16 + v + 8 * half;
      const int lOk = mask[bi * SEQ + l];
      float val = acc[ct][v];
      if (!(mOk && lOk)) val = -1.0e12f;
      if (l > m) val -= 1.0e12f;
      __builtin_nontemporal_store(val * 0.125f,
                                  out + ((size_t)bh * SEQ + l) * SEQ + m);
    }
  }
}

// ---------------------------------------------------------------------------
extern "C" void kernel_launch(void* const* d_in, const int* in_sizes, int n_in,
                              void* d_out, int out_size, void* d_ws, size_t ws_size,
                              hipStream_t stream) {
  const float* X    = (const float*)d_in[0];   // [8,1024,768]
  const float* W    = (const float*)d_in[1];   // [768,1536]
  const float* bias = (const float*)d_in[2];   // [1536]
  const int*   mask = (const int*)d_in[3];     // [8,1024]
  float* out = (float*)d_out;                  // [8,12,1024,1024]

  float* Qws = (float*)d_ws;                              // 6,291,456 floats
  float* Kws = Qws + (size_t)NBATCH * NHEADS * SEQ * HS;  // 6,291,456 floats

  dim3 g1(128, NHEADS);
  qkv_rope_kernel<<<g1, 256, 0, stream>>>(X, W, bias, Qws, Kws);

  dim3 g2(16, 8, NBATCH * NHEADS);
  attn_logits_kernel<<<g2, 256, 0, stream>>>(Qws, Kws, mask, out);
}